// SpGraphAttentionLayer_26164940767560
// MI455X (gfx1250) — compile-verified
//
#include <hip/hip_runtime.h>
#include <hip/hip_bf16.h>
#include <math.h>

typedef __attribute__((ext_vector_type(2))) float v2f;
typedef __attribute__((ext_vector_type(8))) float v8f;

#define N_NODES   100000
#define N_EDGES   1600000
#define IN_F      128
#define ATTN      128
#define NHEADS    4
#define DK        32
#define SLOPE_F   0.2f
#define LDS_STRIDE 132   // 128 + 4 dwords pad -> conflict-free A-frag b64 reads

// ---------------------------------------------------------------------------
// Kernel 1: wx = x @ W with V_WMMA_F32_16X16X4_F32 (fp32 accuracy, wave32).
// Block = 256 threads = 8 waves. Block b owns rows [16b,16b+16); wave w owns
// cols [16w,16w+16) (8 waves x 16 cols = all 128 cols). The 16x128 x-tile is
// staged once in LDS and shared by all 8 waves; W (64KB) stays cache-resident.
// ---------------------------------------------------------------------------
__global__ __launch_bounds__(256) void gat_gemm_wmma(const float* __restrict__ x,
                                                     const float* __restrict__ W,
                                                     float* __restrict__ wx) {
    __shared__ float As[16 * LDS_STRIDE];

    const int tid  = threadIdx.x;
    const int row0 = blockIdx.x * 16;

    // Stage 16x128 floats (512 float4) with 256 threads, 2 float4 each.
    {
        const float4* xv = (const float4*)(x + (size_t)row0 * IN_F);
        #pragma unroll
        for (int it = 0; it < 2; ++it) {
            int idx = tid + it * 256;          // 0..511
            int r   = idx >> 5;                // row 0..15 (32 float4 per row)
            int c4  = idx & 31;                // float4 column
            ((float4*)(As + r * LDS_STRIDE))[c4] = xv[(size_t)r * 32 + c4];
        }
    }
    __syncthreads();

    const int wave = tid >> 5;        // 0..7 -> column tile
    const int lane = tid & 31;
    const int col0 = wave * 16;
    const int m    = lane & 15;       // A-matrix row (both halves)
    const int kb   = (lane >> 4) * 2; // K sub-offset: lanes 0-15 -> K0/1, 16-31 -> K2/3
    const int n    = lane & 15;       // B/C/D column

    v8f c = {};
    const float* Wp = W + col0 + n;   // W is row-major [K=128][N=128]

    #pragma unroll
    for (int k0 = 0; k0 < IN_F; k0 += 4) {
        // A frag (16x4 f32): per ISA 7.12.2 layout
        const float* ap = &As[m * LDS_STRIDE + k0 + kb];
        v2f a; a.x = ap[0]; a.y = ap[1];
        // B frag (4x16 f32): lanes 0-15 rows K0/K1, lanes 16-31 rows K2/K3
        v2f b;
        b.x = Wp[(size_t)(k0 + kb) * ATTN];
        b.y = Wp[(size_t)(k0 + kb + 1) * ATTN];
        c = __builtin_amdgcn_wmma_f32_16x16x4_f32(false, a, false, b,
                                                  (short)0, c, false, false);
    }

    // C/D layout: VGPR r -> M = r + 8*(lane>>4), N = lane&15
    const int rbase = row0 + ((lane >> 4) * 8);
    #pragma unroll
    for (int r = 0; r < 8; ++r) {
        wx[(size_t)(rbase + r) * ATTN + col0 + n] = c[r];
    }
}

// ---------------------------------------------------------------------------
// Kernel 2: per-node head scores alpha_src/alpha_dst; also init m=-inf, den=0.
// ---------------------------------------------------------------------------
__global__ __launch_bounds__(256) void gat_alpha(const float* __restrict__ wx,
                                                 const float* __restrict__ avec,
                                                 float* __restrict__ alpha_src,
                                                 float* __restrict__ alpha_dst,
                                                 float* __restrict__ mmax,
                                                 float* __restrict__ den) {
    int node = blockIdx.x * blockDim.x + threadIdx.x;
    if (node >= N_NODES) return;
    const float* row = wx + (size_t)node * ATTN;
    #pragma unroll
    for (int h = 0; h < NHEADS; ++h) {
        float as = 0.f, ad = 0.f;
        #pragma unroll
        for (int k = 0; k < DK; ++k) {
            float v = row[h * DK + k];
            as = fmaf(v, avec[k], as);
            ad = fmaf(v, avec[DK + k], ad);
        }
        alpha_src[node * NHEADS + h] = as;
        alpha_dst[node * NHEADS + h] = ad;
        mmax[node * NHEADS + h]      = -INFINITY;   // segment_max identity
        den[node * NHEADS + h]       = 0.f;
    }
}

// Float atomic max via int/uint punning (valid with -inf init; handles mixed signs).
__device__ __forceinline__ void atomicMaxF(float* addr, float val) {
    if (val >= 0.f) atomicMax((int*)addr, __float_as_int(val));
    else            atomicMin((unsigned int*)addr, __float_as_uint(val));
}

__device__ __forceinline__ float lrelu(float e) {
    return e > 0.f ? e : SLOPE_F * e;
}

// ---------------------------------------------------------------------------
// Kernel 3: segment max of leaky-ReLU scores over source node (seg = edge[0]).
// ---------------------------------------------------------------------------
__global__ __launch_bounds__(256) void gat_edge_max(const int* __restrict__ edge,
                                                    const float* __restrict__ alpha_src,
                                                    const float* __restrict__ alpha_dst,
                                                    float* __restrict__ mmax) {
    int i = blockIdx.x * blockDim.x + threadIdx.x;
    if (i >= N_EDGES) return;
    int s = edge[i];
    int d = edge[N_EDGES + i];
    #pragma unroll
    for (int h = 0; h < NHEADS; ++h) {
        float e = lrelu(alpha_src[s * NHEADS + h] + alpha_dst[d * NHEADS + h]);
        atomicMaxF(&mmax[s * NHEADS + h], e);
    }
}

// ---------------------------------------------------------------------------
// Kernel 4: num = exp(e - m[seg]); write num into attention slot; den += num.
// (m[seg] is always finite for any seg that appears in an edge, so the
//  reference's isfinite fixup only affects edgeless nodes and drops out.)
// ---------------------------------------------------------------------------
__global__ __launch_bounds__(256) void gat_edge_exp(const int* __restrict__ edge,
                                                    const float* __restrict__ alpha_src,
                                                    const float* __restrict__ alpha_dst,
                                                    const float* __restrict__ mmax,
                                                    float* __restrict__ den,
                                                    float* __restrict__ attn) {
    int i = blockIdx.x * blockDim.x + threadIdx.x;
    if (i >= N_EDGES) return;
    int s = edge[i];
    int d = edge[N_EDGES + i];
    #pragma unroll
    for (int h = 0; h < NHEADS; ++h) {
        float e   = lrelu(alpha_src[s * NHEADS + h] + alpha_dst[d * NHEADS + h]);
        float num = __expf(e - mmax[s * NHEADS + h]);
        attn[(size_t)i * NHEADS + h] = num;
        atomicAdd(&den[s * NHEADS + h], num);
    }
}

// ---------------------------------------------------------------------------
// Kernel 5: attention = num / (den[seg] + 1e-16)
// ---------------------------------------------------------------------------
__global__ __launch_bounds__(256) void gat_edge_norm(const int* __restrict__ edge,
                                                     const float* __restrict__ den,
                                                     float* __restrict__ attn) {
    int i = blockIdx.x * blockDim.x + threadIdx.x;
    if (i >= N_EDGES) return;
    int s = edge[i];
    #pragma unroll
    for (int h = 0; h < NHEADS; ++h) {
        attn[(size_t)i * NHEADS + h] /= (den[s * NHEADS + h] + 1e-16f);
    }
}

extern "C" void kernel_launch(void* const* d_in, const int* in_sizes, int n_in,
                              void* d_out, int out_size, void* d_ws, size_t ws_size,
                              hipStream_t stream) {
    const float* x    = (const float*)d_in[0];
    const int*   edge = (const int*)d_in[1];   // JAX canonicalizes int64 -> int32
    const float* W    = (const float*)d_in[2];
    const float* avec = (const float*)d_in[3]; // [0..31]=a_src, [32..63]=a_dst

    // Output tuple (attention, wx) flattened in return order.
    float* attn = (float*)d_out;                                   // N_EDGES*4
    float* wx   = (float*)d_out + (size_t)N_EDGES * NHEADS;        // N_NODES*128

    // Workspace: alpha_src | alpha_dst | mmax | den  (4 * 400000 floats = 6.4MB)
    float* alpha_src = (float*)d_ws;
    float* alpha_dst = alpha_src + (size_t)N_NODES * NHEADS;
    float* mmax      = alpha_dst + (size_t)N_NODES * NHEADS;
    float* den       = mmax      + (size_t)N_NODES * NHEADS;

    (void)in_sizes; (void)n_in; (void)out_size; (void)ws_size;

    gat_gemm_wmma<<<N_NODES / 16, 256, 0, stream>>>(x, W, wx);            // 6250 blocks
    gat_alpha<<<(N_NODES + 255) / 256, 256, 0, stream>>>(wx, avec, alpha_src,
                                                         alpha_dst, mmax, den);
    gat_edge_max<<<(N_EDGES + 255) / 256, 256, 0, stream>>>(edge, alpha_src,
                                                            alpha_dst, mmax);
    gat_edge_exp<<<(N_EDGES + 255) / 256, 256, 0, stream>>>(edge, alpha_src,
                                                            alpha_dst, mmax, den, attn);
    gat_edge_norm<<<(N_EDGES + 255) / 256, 256, 0, stream>>>(edge, den, attn);
}